// STGS_67207648248400
// MI455X (gfx1250) — compile-verified
//
#include <hip/hip_runtime.h>

// Straight-through Gumbel-softmax, one workgroup per vocab row.
// rows = B*S = 2048, V = 32000.  ~1.31 GB HBM traffic -> ~56 us floor @ 23.3 TB/s.
//
// Phase 1: async-stage x,u,u2 chunks into LDS (global_load_async_to_lds_b128,
//          double-buffered per wave, s_wait_asynccnt pipeline), compute
//          logits = x + gumbel(u) with v_log_f32 fast logs, cache logits in LDS,
//          track row max and argmax(logits + gumbel(u2)).
// Phase 2: e = __expf(logit - M) in-place in LDS, accumulate sum.
// Phase 3: write y_soft = e/S and hard one-hot (lane-contiguous b32 stores;
//          the y_soft slice of d_out is not 16B-aligned).

#define VOCAB     32000
#define NTHREADS  256
#define CHUNK     1024        // NTHREADS * 4 floats per chunk
#define NFULL     31          // 31 * 1024 = 31744
#define TAIL_BASE 31744       // last 256 elements -> 1 per thread
#define NWAVES    8

// One b128 async copy: each lane moves 16B  global -> its own LDS slot.
// VDST = per-lane LDS byte address (low 32 bits of generic pointer),
// VADDR = per-lane 64-bit global address (GV mode).
__device__ __forceinline__ void async_b128(unsigned lds_addr, const void* gaddr) {
    asm volatile("global_load_async_to_lds_b128 %0, %1, off"
                 :: "v"(lds_addr), "v"(gaddr)
                 : "memory");
}

__global__ __launch_bounds__(NTHREADS)
void stgs_row_kernel(const float* __restrict__ x,
                     const float* __restrict__ u,
                     const float* __restrict__ u2,
                     float* __restrict__ out_ids,     // [rows]
                     float* __restrict__ out_onehot,  // [rows*V]
                     float* __restrict__ out_temp,    // [1]
                     float* __restrict__ out_ysoft)   // [rows*V]
{
    __shared__ __align__(16) float s_logits[VOCAB];        // 128000 B row cache
    __shared__ __align__(16) float s_stage[2][3][CHUNK];   // 24576 B async staging
    __shared__ float red_m[NWAVES], red_v[NWAVES];
    __shared__ int   red_i[NWAVES];
    __shared__ float s_M, s_invS;
    __shared__ int   s_bi;

    const int    row  = blockIdx.x;
    const int    tid  = threadIdx.x;
    const size_t base = (size_t)row * VOCAB;
    const float* __restrict__ xr  = x  + base;
    const float* __restrict__ ur  = u  + base;
    const float* __restrict__ u2r = u2 + base;

    const float A = 0.999f;     // 0.999 - 1e-12 rounds to 0.999f in fp32
    const float E = 1e-12f;

    // per-lane LDS byte address of this thread's staging slot in buffer 0:
    // generic LDS pointers carry the LDS offset in their low 32 bits.
    const unsigned lds0 = (unsigned)(uintptr_t)&s_stage[0][0][tid * 4];
    // array stride 4096 B, buffer stride 12288 B

    float mmax = -INFINITY;     // row max of logits
    float bv   = -INFINITY;     // argmax value of logits + g2
    int   bi   = 0;

    // prime the pipeline: chunk 0 -> buffer 0 (3 async b128 per wave-lane)
    {
        const int off = tid * 4;
        async_b128(lds0,          xr  + off);
        async_b128(lds0 + 4096u,  ur  + off);
        async_b128(lds0 + 8192u,  u2r + off);
    }

    for (int i = 0; i < NFULL; ++i) {
        const int buf = i & 1;
        if (i + 1 < NFULL) {
            const int      noff = (i + 1) * CHUNK + tid * 4;
            const unsigned nlds = lds0 + (unsigned)(((i + 1) & 1) * 12288);
            async_b128(nlds,          xr  + noff);
            async_b128(nlds + 4096u,  ur  + noff);
            async_b128(nlds + 8192u,  u2r + noff);
            asm volatile("s_wait_asynccnt 0x3" ::: "memory"); // chunk i complete
        } else {
            asm volatile("s_wait_asynccnt 0x0" ::: "memory");
        }

        const int    off = i * CHUNK + tid * 4;
        const float4 xv  = *(const float4*)&s_stage[buf][0][tid * 4];
        const float4 uv  = *(const float4*)&s_stage[buf][1][tid * 4];
        const float4 u2v = *(const float4*)&s_stage[buf][2][tid * 4];
        const float xs[4]  = {xv.x,  xv.y,  xv.z,  xv.w};
        const float uu[4]  = {uv.x,  uv.y,  uv.z,  uv.w};
        const float uu2[4] = {u2v.x, u2v.y, u2v.z, u2v.w};
        #pragma unroll
        for (int k = 0; k < 4; ++k) {
            const float g = -__logf(-__logf(fmaf(uu[k], A, E)));   // v_log_f32 x2
            const float l = xs[k] + g;
            s_logits[off + k] = l;
            mmax = fmaxf(mmax, l);
            const float g2 = -__logf(-__logf(fmaf(uu2[k], A, E)));
            const float t  = l + g2;
            if (t > bv) { bv = t; bi = off + k; }   // strict '>' keeps first max
        }
    }
    { // tail: 256 scalar elements via plain loads
        const int   off = TAIL_BASE + tid;
        const float l   = xr[off] - __logf(-__logf(fmaf(ur[off], A, E)));
        s_logits[off] = l;
        mmax = fmaxf(mmax, l);
        const float t = l - __logf(-__logf(fmaf(u2r[off], A, E)));
        if (t > bv) { bv = t; bi = off; }
    }

    // wave32 butterfly: max + index-aware argmax
    #pragma unroll
    for (int d = 16; d > 0; d >>= 1) {
        mmax = fmaxf(mmax, __shfl_xor(mmax, d, 32));
        const float v2 = __shfl_xor(bv, d, 32);
        const int   i2 = __shfl_xor(bi, d, 32);
        if (v2 > bv || (v2 == bv && i2 < bi)) { bv = v2; bi = i2; }
    }
    const int wave = tid >> 5;
    if ((tid & 31) == 0) { red_m[wave] = mmax; red_v[wave] = bv; red_i[wave] = bi; }
    __syncthreads();
    if (tid == 0) {
        float M = red_m[0], BV = red_v[0];
        int   BI = red_i[0];
        #pragma unroll
        for (int w = 1; w < NWAVES; ++w) {
            M = fmaxf(M, red_m[w]);
            const float v2 = red_v[w]; const int i2 = red_i[w];
            if (v2 > BV || (v2 == BV && i2 < BI)) { BV = v2; BI = i2; }
        }
        s_M = M; s_bi = BI;
        out_ids[row] = (float)BI;            // diff_output_ids == ids numerically
        if (row == 0) out_temp[0] = 1.0f;
    }
    __syncthreads();
    const float M = s_M;

    // Phase 2: exponentiate in place (own slots only), accumulate sum
    float acc = 0.0f;
    for (int i = 0; i < NFULL; ++i) {
        const int    o  = i * CHUNK + tid * 4;
        const float4 l4 = *(const float4*)&s_logits[o];
        float4 e4;
        e4.x = __expf(l4.x - M);
        e4.y = __expf(l4.y - M);
        e4.z = __expf(l4.z - M);
        e4.w = __expf(l4.w - M);
        *(float4*)&s_logits[o] = e4;
        acc += (e4.x + e4.y) + (e4.z + e4.w);
    }
    {
        const int   o = TAIL_BASE + tid;
        const float e = __expf(s_logits[o] - M);
        s_logits[o] = e;
        acc += e;
    }
    #pragma unroll
    for (int d = 16; d > 0; d >>= 1) acc += __shfl_xor(acc, d, 32);
    if ((tid & 31) == 0) red_m[wave] = acc;     // red_m reusable now
    __syncthreads();
    if (tid == 0) {
        float S = 0.0f;
        #pragma unroll
        for (int w = 0; w < NWAVES; ++w) S += red_m[w];
        s_invS = 1.0f / S;
    }
    __syncthreads();

    const float invS = s_invS;
    const int   bidx = s_bi;
    float* __restrict__ oh = out_onehot + base;
    float* __restrict__ ys = out_ysoft  + base;
    // lane-contiguous b32 writes (y_soft slice base in d_out is 4B-aligned only)
    for (int i = 0; i < NFULL; ++i) {
        const int off0 = i * CHUNK + tid;
        #pragma unroll
        for (int k = 0; k < 4; ++k) {
            const int   off = off0 + k * NTHREADS;
            const float e   = s_logits[off];
            ys[off] = e * invS;
            oh[off] = (off == bidx) ? 1.0f : 0.0f;
        }
    }
    {
        const int   off = TAIL_BASE + tid;
        const float e   = s_logits[off];
        ys[off] = e * invS;
        oh[off] = (off == bidx) ? 1.0f : 0.0f;
    }
}

extern "C" void kernel_launch(void* const* d_in, const int* in_sizes, int n_in,
                              void* d_out, int out_size, void* d_ws, size_t ws_size,
                              hipStream_t stream) {
    const float* x  = (const float*)d_in[0];
    const float* u  = (const float*)d_in[1];
    const float* u2 = (const float*)d_in[2];

    const int    rows  = in_sizes[0] / VOCAB;        // B*S = 2048
    const size_t nelem = (size_t)rows * VOCAB;       // 65,536,000

    float* out        = (float*)d_out;
    float* out_ids    = out;                         // [rows]
    float* out_onehot = out + rows;                  // [rows*V]
    float* out_temp   = out + rows + nelem;          // [1]
    float* out_ysoft  = out_temp + 1;                // [rows*V]

    stgs_row_kernel<<<dim3(rows), dim3(NTHREADS), 0, stream>>>(
        x, u, u2, out_ids, out_onehot, out_temp, out_ysoft);
}